// TurbineGNN_90022514524788
// MI455X (gfx1250) — compile-verified
//
#include <hip/hip_runtime.h>
#include <hip/hip_bf16.h>

#define N_NODES 100000
#define N_EDGES 3200000

typedef __attribute__((ext_vector_type(2))) float v2f;
typedef __attribute__((ext_vector_type(8))) float v8f;

__device__ __forceinline__ void atomic_add_f32(float* p, float v) {
    (void)__hip_atomic_fetch_add(p, v, __ATOMIC_RELAXED, __HIP_MEMORY_SCOPE_AGENT);
}

// ---------------------------------------------------------------------------
// Degree / normalization (shared by all three layers)
// ---------------------------------------------------------------------------
__global__ void __launch_bounds__(256) init_deg(float* __restrict__ deg) {
    int n = blockIdx.x * 256 + threadIdx.x;
    if (n < N_NODES) deg[n] = 1.0f;  // self-loop
}

__global__ void __launch_bounds__(256) count_deg(const int* __restrict__ cols,
                                                 float* __restrict__ deg) {
    int e = blockIdx.x * 256 + threadIdx.x;
    if (e < N_EDGES) atomic_add_f32(&deg[cols[e]], 1.0f);
}

__global__ void __launch_bounds__(256) deg_to_dinv(float* __restrict__ deg) {
    int n = blockIdx.x * 256 + threadIdx.x;
    if (n < N_NODES) deg[n] = rsqrtf(deg[n]);  // deg >= 1 always (self-loop)
}

// ---------------------------------------------------------------------------
// Dense GEMM via V_WMMA_F32_16X16X4_F32.
// One block = 16 output rows; one wave = one 16-column tile.
// H row-tile and full W staged in LDS; ReLU fused into the H load.
// ---------------------------------------------------------------------------
template<int KIN, int KOUT, bool RELU>
__global__ void __launch_bounds__(32 * (KOUT / 16))
gcn_gemm(const float* __restrict__ H, const float* __restrict__ W,
         float* __restrict__ XW) {
    __shared__ float lH[16 * KIN];
    __shared__ float lW[KIN * KOUT];

    const int tid = threadIdx.x;
    const int nthreads = 32 * (KOUT / 16);
    const int m0 = blockIdx.x * 16;

    for (int i = tid; i < 16 * KIN; i += nthreads) {
        float v = H[(m0 + i / KIN) * KIN + (i % KIN)];
        if (RELU) v = fmaxf(v, 0.0f);
        lH[i] = v;
    }
    for (int i = tid; i < KIN * KOUT; i += nthreads) lW[i] = W[i];
    __syncthreads();

    const int wave = tid >> 5;
    const int lane = tid & 31;
    const int half = lane >> 4;   // 0: K pair {0,1}; 1: K pair {2,3}
    const int l    = lane & 15;
    const int n0   = wave * 16;   // column-tile base

    v8f acc = {};
#pragma unroll 4
    for (int k = 0; k < KIN; k += 4) {
        const int kA = k + 2 * half;
        v2f a, b;
        // A 16x4 f32 layout: lanes 0-15 hold K={0,1}, lanes 16-31 hold K={2,3}
        a.x = lH[l * KIN + kA];
        a.y = lH[l * KIN + kA + 1];
        // B 4x16 f32 layout mirrors A: row K striped across lanes within a VGPR
        b.x = lW[kA * KOUT + n0 + l];
        b.y = lW[(kA + 1) * KOUT + n0 + l];
        acc = __builtin_amdgcn_wmma_f32_16x16x4_f32(
            /*neg_a=*/false, a, /*neg_b=*/false, b,
            /*c_mod=*/(short)0, acc, /*reuse_a=*/false, /*reuse_b=*/false);
    }

    // C/D layout: VGPR r -> M = r (lanes 0-15) / M = 8+r (lanes 16-31)
#pragma unroll
    for (int r = 0; r < 8; ++r) {
        XW[(m0 + half * 8 + r) * KOUT + n0 + l] = acc[r];
    }
}

// ---------------------------------------------------------------------------
// Aggregation: h[n] = xw[n]*dinv[n]^2 + bias (self-loop term), then edge
// scatter h[col] += xw[row] * dinv[row] * dinv[col].
// ---------------------------------------------------------------------------
template<int F>
__global__ void __launch_bounds__(256) agg_init(const float* __restrict__ xw,
                                                const float* __restrict__ dinv,
                                                const float* __restrict__ bias,
                                                float* __restrict__ h) {
    int idx = blockIdx.x * 256 + threadIdx.x;
    if (idx >= N_NODES * F) return;
    int n = idx / F, f = idx % F;
    float di = dinv[n];
    h[idx] = xw[idx] * di * di + bias[f];
}

template<int F>
__global__ void __launch_bounds__(256) agg_scatter(const float* __restrict__ xw,
                                                   const int* __restrict__ rows,
                                                   const int* __restrict__ cols,
                                                   const float* __restrict__ dinv,
                                                   float* __restrict__ h) {
    const int epb = 256 / F;  // edges per block
    int e = blockIdx.x * epb + threadIdx.x / F;
    int f = threadIdx.x % F;
    if (e >= N_EDGES) return;
    __builtin_prefetch(&rows[e + 32 * epb], 0, 0);  // global_prefetch_b8
    int r = rows[e];
    int c = cols[e];
    float norm = dinv[r] * dinv[c];          // scalar-uniform within the F-lane group
    float v = xw[r * F + f] * norm;          // coalesced 4B*F contiguous gather
    atomic_add_f32(&h[c * F + f], v);        // global_atomic_add_f32 at L2
}

// ---------------------------------------------------------------------------
// Final projection: out[n] = sum_k relu(h3[n,k]) * Wp[k] + bp
// ---------------------------------------------------------------------------
__global__ void __launch_bounds__(256) final_proj(const float* __restrict__ h3,
                                                  const float* __restrict__ Wp,
                                                  const float* __restrict__ bp,
                                                  float* __restrict__ out) {
    int n = blockIdx.x * 256 + threadIdx.x;
    if (n >= N_NODES) return;
    float s = bp[0];
#pragma unroll
    for (int k = 0; k < 32; ++k)
        s += fmaxf(h3[n * 32 + k], 0.0f) * Wp[k];
    out[n] = s;
}

// ---------------------------------------------------------------------------
extern "C" void kernel_launch(void* const* d_in, const int* in_sizes, int n_in,
                              void* d_out, int out_size, void* d_ws, size_t ws_size,
                              hipStream_t stream) {
    const float* x  = (const float*)d_in[0];
    const int*   ei = (const int*)d_in[1];
    const float* W1 = (const float*)d_in[2];
    const float* b1 = (const float*)d_in[3];
    const float* W2 = (const float*)d_in[4];
    const float* b2 = (const float*)d_in[5];
    const float* W3 = (const float*)d_in[6];
    const float* b3 = (const float*)d_in[7];
    const float* Wp = (const float*)d_in[8];
    const float* bp = (const float*)d_in[9];
    float* out = (float*)d_out;

    const int* rows = ei;             // edge_index[0] = source
    const int* cols = ei + N_EDGES;   // edge_index[1] = target

    // Workspace: dinv (0.4MB) | bufA (25.6MB) | bufB (25.6MB)
    char* ws = (char*)d_ws;
    const size_t offA = ((size_t)N_NODES * 4 + 255) & ~(size_t)255;
    const size_t offB = offA + (((size_t)N_NODES * 64 * 4 + 255) & ~(size_t)255);
    float* dinv = (float*)ws;
    float* bufA = (float*)(ws + offA);
    float* bufB = (float*)(ws + offB);

    const int nB = (N_NODES + 255) / 256;
    const int eB = (N_EDGES + 255) / 256;
    const int mtiles = N_NODES / 16;  // 6250, exact

    // Shared normalization (deg -> rsqrt) once for all layers
    init_deg<<<nB, 256, 0, stream>>>(dinv);
    count_deg<<<eB, 256, 0, stream>>>(cols, dinv);
    deg_to_dinv<<<nB, 256, 0, stream>>>(dinv);

    // Layer 1: x(128) -> 64
    gcn_gemm<128, 64, false><<<mtiles, 128, 0, stream>>>(x, W1, bufA);
    agg_init<64><<<(N_NODES * 64 + 255) / 256, 256, 0, stream>>>(bufA, dinv, b1, bufB);
    agg_scatter<64><<<(N_EDGES + 3) / 4, 256, 0, stream>>>(bufA, rows, cols, dinv, bufB);

    // Layer 2: relu(h1)(64) -> 64
    gcn_gemm<64, 64, true><<<mtiles, 128, 0, stream>>>(bufB, W2, bufA);
    agg_init<64><<<(N_NODES * 64 + 255) / 256, 256, 0, stream>>>(bufA, dinv, b2, bufB);
    agg_scatter<64><<<(N_EDGES + 3) / 4, 256, 0, stream>>>(bufA, rows, cols, dinv, bufB);

    // Layer 3: relu(h2)(64) -> 32
    gcn_gemm<64, 32, true><<<mtiles, 64, 0, stream>>>(bufB, W3, bufA);
    agg_init<32><<<(N_NODES * 32 + 255) / 256, 256, 0, stream>>>(bufA, dinv, b3, bufB);
    agg_scatter<32><<<(N_EDGES + 7) / 8, 256, 0, stream>>>(bufA, rows, cols, dinv, bufB);

    // Final: relu(h3) @ Wp + bp
    final_proj<<<nB, 256, 0, stream>>>(bufB, Wp, bp, out);
}